// GraphConvolution_69758858822454
// MI455X (gfx1250) — compile-verified
//
#include <hip/hip_runtime.h>
#include <math.h>

typedef __attribute__((ext_vector_type(2))) float v2f;
typedef __attribute__((ext_vector_type(8))) float v8f;

// ---------------------------------------------------------------- utilities
__global__ void fill_zero_f32(float* __restrict__ p, long long n) {
  long long i = (long long)blockIdx.x * blockDim.x + threadIdx.x;
  long long stride = (long long)gridDim.x * blockDim.x;
  for (; i < n; i += stride) p[i] = 0.0f;
}

// deg = segment_sum(ones, src)
__global__ void degree_kernel(const int* __restrict__ src,
                              float* __restrict__ deg, int E) {
  int i = blockIdx.x * blockDim.x + threadIdx.x;
  int stride = gridDim.x * blockDim.x;
  for (; i < E; i += stride) atomicAdd(deg + src[i], 1.0f);
}

// dinv = deg>0 ? 1/sqrt(deg) : 0
__global__ void dinv_kernel(const float* __restrict__ deg,
                            float* __restrict__ dinv, int N) {
  int i = blockIdx.x * blockDim.x + threadIdx.x;
  if (i < N) {
    float d = deg[i];
    dinv[i] = (d > 0.0f) ? (1.0f / sqrtf(d)) : 0.0f;
  }
}

// ------------------------------------------------- W fragment pre-pack
// Wp laid out so each wave's per-step B fragment is one contiguous b64:
//   Wp[((t*32 + s)*32 + lane)*2 ..] = { W[4s+hi][16t+l16], W[4s+2+hi][16t+l16] }
// t = column tile (0..7), s = K step (0..31), lane = 0..31.
__global__ void pack_w_kernel(const float* __restrict__ W,
                              float* __restrict__ Wp) {
  int i = blockIdx.x * blockDim.x + threadIdx.x;   // 0..8191
  if (i < 8 * 32 * 32) {
    int lane = i & 31;
    int s    = (i >> 5) & 31;
    int t    = i >> 10;
    int l16  = lane & 15;
    int hi   = lane >> 4;
    int col  = t * 16 + l16;
    Wp[2 * i + 0] = W[(s * 4 + hi) * 128 + col];
    Wp[2 * i + 1] = W[(s * 4 + 2 + hi) * 128 + col];
  }
}

// ------------------------------------------------------------ WMMA fp32 GEMM
// XW[N,128] = X[N,128] @ W[128,128].
// Block = 256 threads = 8 waves; block owns a 16-row tile, wave w owns
// column tile w. A tile staged in LDS (padded rows -> bank-conflict-free
// ds_load_2addr_b64), B from pre-packed fragments (one global_load_b64/step).
#define A_PITCH 132   // 128 + 4 floats padding
__global__ __launch_bounds__(256) void gemm_xw_wmma(
    const float* __restrict__ X, const float* __restrict__ Wp,
    float* __restrict__ XW, int nRows) {
  __shared__ float As[16 * A_PITCH];

  const int tid  = threadIdx.x;
  const int wave = tid >> 5;          // 0..7 -> column tile
  const int lane = tid & 31;
  const int l16  = lane & 15;
  const int hi   = lane >> 4;
  const int rowBase = blockIdx.x * 16;

  // Cooperative A-tile load: thread t loads 8 floats of row (t/16).
  // Rows past nRows are clamped (duplicates only feed masked-out C rows).
  {
    int r  = tid >> 4;                // 0..15
    int cc = (tid & 15) * 8;          // 0..120
    int gr = rowBase + r;
    if (gr >= nRows) gr = nRows - 1;
    const float4* s4 = (const float4*)(X + (long long)gr * 128 + cc);
    float4 v0 = s4[0];
    float4 v1 = s4[1];
    float* dp = As + r * A_PITCH + cc;
    ((float4*)dp)[0] = v0;
    ((float4*)dp)[1] = v1;
  }
  __syncthreads();

  const float* wp = Wp + (long long)wave * 32 * 64;  // this col tile's frags
  const float* ap = As + l16 * A_PITCH + hi * 2;

  v8f c = {};
#pragma unroll 4
  for (int s = 0; s < 32; ++s) {
    v2f a = *(const v2f*)(ap + s * 4);             // ds_load, conflict-free
    v2f b = *(const v2f*)(wp + s * 64 + lane * 2); // contiguous global b64
    // 8 args: (neg_a, A, neg_b, B, c_mod, C, reuse_a, reuse_b)
    c = __builtin_amdgcn_wmma_f32_16x16x4_f32(false, a, false, b,
                                              (short)0, c, false, false);
  }

  // C/D layout: VGPR r -> M = r (lanes 0-15) / r+8 (lanes 16-31), N = lane%16
  const int colBase = wave * 16;
  if (rowBase + 16 <= nRows) {
    // full tile: wave-uniform fast path, no exec-mask juggling
    float* op = XW + (long long)(rowBase + hi * 8) * 128 + colBase + l16;
#pragma unroll
    for (int r = 0; r < 8; ++r) {
      op[(long long)r * 128] = c[r];
    }
  } else {
    // ragged tail block only
#pragma unroll
    for (int r = 0; r < 8; ++r) {
      int mm = rowBase + r + hi * 8;
      if (mm < nRows) XW[(long long)mm * 128 + colBase + l16] = c[r];
    }
  }
}

// ------------------------------------------------- edge gather/scale/scatter
// One wave per edge: lane handles 4 consecutive units (128/32 = 4).
__global__ __launch_bounds__(256) void scatter_edges(
    const int* __restrict__ src, const int* __restrict__ dst,
    const float* __restrict__ dinv, const float* __restrict__ xw,
    float* __restrict__ out, int E) {
  const int lane = threadIdx.x & 31;
  int w = (int)((blockIdx.x * (long long)blockDim.x + threadIdx.x) >> 5);
  const int nWaves = (int)(((long long)gridDim.x * blockDim.x) >> 5);
  for (int e = w; e < E; e += nWaves) {
    // e is wave-uniform: force the edge endpoints into SGPRs
    const int s = __builtin_amdgcn_readfirstlane(src[e]);
    const int d = __builtin_amdgcn_readfirstlane(dst[e]);
    const float coef = dinv[s] * dinv[d];
    if (coef != 0.0f) {
      const float4 v = ((const float4*)(xw + (long long)d * 128))[lane];
      float* orow = out + (long long)s * 128 + lane * 4;
      atomicAdd(orow + 0, v.x * coef);
      atomicAdd(orow + 1, v.y * coef);
      atomicAdd(orow + 2, v.z * coef);
      atomicAdd(orow + 3, v.w * coef);
    }
  }
}

// ------------------------------------------------------------- bias + relu
__global__ void bias_relu_kernel(float* __restrict__ out,
                                 const float* __restrict__ bias,
                                 long long total) {
  long long i = (long long)blockIdx.x * blockDim.x + threadIdx.x;
  long long stride = (long long)gridDim.x * blockDim.x;
  for (; i < total; i += stride) {
    out[i] = fmaxf(out[i] + bias[i & 127], 0.0f);
  }
}

// ---------------------------------------------------------------- launcher
extern "C" void kernel_launch(void* const* d_in, const int* in_sizes, int n_in,
                              void* d_out, int out_size, void* d_ws, size_t ws_size,
                              hipStream_t stream) {
  const float* node_states = (const float*)d_in[0];
  const int*   src         = (const int*)d_in[1];
  const int*   dst         = (const int*)d_in[2];
  const float* kernel      = (const float*)d_in[3];
  const float* bias        = (const float*)d_in[4];

  const int UNITS = in_sizes[4];             // 128
  const int F     = in_sizes[3] / UNITS;     // 128
  const int N     = in_sizes[0] / F;         // 100000
  const int E     = in_sizes[1];             // 1600000

  float* out = (float*)d_out;

  // workspace layout: deg[N] | dinv[N] | Wp[8*32*32*2] | xw[N*UNITS]
  float* deg  = (float*)d_ws;
  float* dinv = deg + N;
  float* Wp   = dinv + N;
  float* xw   = Wp + 8 * 32 * 32 * 2;

  // 1) zero accumulators (harness poisons buffers; must re-zero every call)
  fill_zero_f32<<<1024, 256, 0, stream>>>(deg, (long long)N);
  fill_zero_f32<<<4096, 256, 0, stream>>>(out, (long long)N * UNITS);

  // 2) degrees
  degree_kernel<<<2048, 256, 0, stream>>>(src, deg, E);

  // 3) dinv
  dinv_kernel<<<(N + 255) / 256, 256, 0, stream>>>(deg, dinv, N);

  // 4a) pre-pack W into WMMA B-fragment order
  pack_w_kernel<<<32, 256, 0, stream>>>(kernel, Wp);

  // 4b) XW = X @ W  (fp32 WMMA)
  gemm_xw_wmma<<<(N + 15) / 16, 256, 0, stream>>>(node_states, Wp, xw, N);

  // 5) edge messages + scatter-add (wave per edge)
  scatter_edges<<<8192, 256, 0, stream>>>(src, dst, dinv, xw, out, E);

  // 6) bias + relu
  bias_relu_kernel<<<4096, 256, 0, stream>>>(out, bias, (long long)N * UNITS);
}